// GCN_9629316678064
// MI455X (gfx1250) — compile-verified
//
#include <hip/hip_runtime.h>
#include <hip/hip_bf16.h>

typedef __attribute__((ext_vector_type(2))) float v2f;
typedef __attribute__((ext_vector_type(8))) float v8f;

#define HIDDEN 64
#define N_OUT_CLS 10

// ---------------------------------------------------------------------------
// utility: fill a float buffer with a constant
// ---------------------------------------------------------------------------
__global__ void fill_kernel(float* __restrict__ p, float v, long long n) {
  long long i = (long long)blockIdx.x * blockDim.x + threadIdx.x;
  if (i < n) p[i] = v;
}

// ---------------------------------------------------------------------------
// degree accumulation: deg[dst[e]] += 1  (deg pre-filled with 1.0 = self loop)
// ---------------------------------------------------------------------------
__global__ void deg_kernel(const int* __restrict__ dst, float* __restrict__ deg,
                           int n_edges) {
  int e = blockIdx.x * blockDim.x + threadIdx.x;
  if (e < n_edges) atomicAdd(&deg[dst[e]], 1.0f);
}

__global__ void dinv_kernel(float* __restrict__ deg, int n) {
  int i = blockIdx.x * blockDim.x + threadIdx.x;
  if (i < n) deg[i] = rsqrtf(deg[i]);
}

// ---------------------------------------------------------------------------
// WMMA GEMM: C[M,64] = A[M,K] * B[K,64], row-major, f32 via v_wmma_f32_16x16x4_f32
// block = 256 threads (8 waves). Each wave computes one 16x16 tile of C.
// Block covers 2 M-tiles x 4 N-tiles. Requires M % 32 == 0 (true: 100000).
// ---------------------------------------------------------------------------
template <int K>
__global__ void wmma_gemm_kernel(const float* __restrict__ A,
                                 const float* __restrict__ B,
                                 float* __restrict__ C, int M) {
  const int wave = threadIdx.x >> 5;
  const int lane = threadIdx.x & 31;
  const int mtile = blockIdx.x * 2 + (wave >> 2);
  const int ntile = wave & 3;                    // HIDDEN/16 == 4
  const int m0 = mtile * 16;
  if (m0 >= M) return;                           // whole-wave exit; EXEC stays full otherwise
  const int n0 = ntile * 16;
  const int lm = lane & 15;                      // row/col within half-wave
  const int hi = lane >> 4;                      // 0 -> K pair {0,1}, 1 -> K pair {2,3}

  const float* __restrict__ arow = A + (long long)(m0 + lm) * K;   // A row for this lane
  const float* __restrict__ bcol = B + (n0 + lm);                  // B column for this lane

  v8f acc = {};
#pragma unroll
  for (int k0 = 0; k0 < K; k0 += 4) {
    v2f a, b;
    // 32-bit A-matrix 16x4 layout: VGPR0 = K{0|2}, VGPR1 = K{1|3} split by lane half
    a.x = arow[k0 + 2 * hi + 0];
    a.y = arow[k0 + 2 * hi + 1];
    // B (4x16) mirrored layout: rows striped across lanes within each VGPR
    b.x = bcol[(long long)(k0 + 2 * hi + 0) * HIDDEN];
    b.y = bcol[(long long)(k0 + 2 * hi + 1) * HIDDEN];
    acc = __builtin_amdgcn_wmma_f32_16x16x4_f32(false, a, false, b,
                                                (short)0, acc, false, false);
  }

  // C/D layout: VGPR r, lanes 0-15 -> M = m0 + r,     N = n0 + lane
  //                     lanes 16-31 -> M = m0 + 8 + r, N = n0 + lane - 16
  float* __restrict__ crow = C + (long long)(m0 + 8 * hi) * HIDDEN + n0 + lm;
#pragma unroll
  for (int r = 0; r < 8; ++r) crow[(long long)r * HIDDEN] = acc[r];
}

// ---------------------------------------------------------------------------
// Edge scatter: agg[dst[e], :] += dinv[src[e]]*dinv[dst[e]] * xw[src[e], :]
// 16 threads per edge, float4 gather, 4x global_atomic_add_f32 scatter.
// ---------------------------------------------------------------------------
__global__ void scatter_kernel(const float* __restrict__ xw,
                               const int* __restrict__ src,
                               const int* __restrict__ dst,
                               const float* __restrict__ dinv,
                               float* __restrict__ agg, long long n_edges) {
  long long t = (long long)blockIdx.x * blockDim.x + threadIdx.x;
  long long e = t >> 4;
  if (e >= n_edges) return;
  int c4 = (int)(t & 15);
  int s = src[e];
  int d = dst[e];
  float nrm = dinv[s] * dinv[d];
  const float4 v = *(const float4*)(xw + (long long)s * HIDDEN + c4 * 4);
  float* out = agg + (long long)d * HIDDEN + c4 * 4;
  atomicAdd(out + 0, nrm * v.x);
  atomicAdd(out + 1, nrm * v.y);
  atomicAdd(out + 2, nrm * v.z);
  atomicAdd(out + 3, nrm * v.w);
}

// ---------------------------------------------------------------------------
// Fused self-loop + bias (+ optional ReLU):
//   h = agg + dinv^2 * xw + bias ; relu optionally
// ---------------------------------------------------------------------------
__global__ void fuse_kernel(const float* __restrict__ agg,
                            const float* __restrict__ xw,
                            const float* __restrict__ dinv,
                            const float* __restrict__ bias,
                            float* __restrict__ out, int n, int do_relu) {
  long long t = (long long)blockIdx.x * blockDim.x + threadIdx.x;
  int i = (int)(t >> 6);
  if (i >= n) return;
  int c = (int)(t & 63);
  float di = dinv[i];
  float v = agg[t] + di * di * xw[t] + bias[c];
  out[t] = do_relu ? fmaxf(v, 0.0f) : v;
}

// ---------------------------------------------------------------------------
// Global mean pool (accumulate): pooled[batch[i], :] += h[i, :]; cnt[batch[i]]++
// ---------------------------------------------------------------------------
__global__ void pool_kernel(const float* __restrict__ h,
                            const int* __restrict__ batch,
                            float* __restrict__ pooled, float* __restrict__ cnt,
                            int n) {
  long long t = (long long)blockIdx.x * blockDim.x + threadIdx.x;
  int i = (int)(t >> 6);
  if (i >= n) return;
  int c = (int)(t & 63);
  int g = batch[i];
  atomicAdd(&pooled[(long long)g * HIDDEN + c], h[t]);
  if (c == 0) atomicAdd(&cnt[g], 1.0f);
}

// ---------------------------------------------------------------------------
// Head: out[g, o] = (pooled[g,:] / max(cnt[g],1)) @ Wlin[:, o] + blin[o]
// ---------------------------------------------------------------------------
__global__ void head_kernel(const float* __restrict__ pooled,
                            const float* __restrict__ cnt,
                            const float* __restrict__ Wlin,
                            const float* __restrict__ blin,
                            float* __restrict__ out, int n_graphs) {
  int t = blockIdx.x * blockDim.x + threadIdx.x;
  if (t >= n_graphs * N_OUT_CLS) return;
  int g = t / N_OUT_CLS;
  int o = t % N_OUT_CLS;
  float inv = 1.0f / fmaxf(cnt[g], 1.0f);
  float acc = blin[o];
#pragma unroll
  for (int h = 0; h < HIDDEN; ++h)
    acc += (pooled[(long long)g * HIDDEN + h] * inv) * Wlin[h * N_OUT_CLS + o];
  out[t] = acc;
}

// ---------------------------------------------------------------------------
// Orchestration
// ---------------------------------------------------------------------------
extern "C" void kernel_launch(void* const* d_in, const int* in_sizes, int n_in,
                              void* d_out, int out_size, void* d_ws, size_t ws_size,
                              hipStream_t stream) {
  const float* x      = (const float*)d_in[0];
  const int*   edges  = (const int*)d_in[1];
  const int*   batch  = (const int*)d_in[2];
  const float* W1     = (const float*)d_in[3];
  const float* b1     = (const float*)d_in[4];
  const float* W2     = (const float*)d_in[5];
  const float* b2     = (const float*)d_in[6];
  const float* Wlin   = (const float*)d_in[7];
  const float* blin   = (const float*)d_in[8];
  float* out = (float*)d_out;

  const int n_feat   = 128;
  const int n_nodes  = in_sizes[0] / n_feat;          // 100000
  const long long n_edges = in_sizes[1] / 2;          // 3200000
  const int n_graphs = out_size / N_OUT_CLS;          // 128

  const int* src = edges;
  const int* dst = edges + n_edges;

  // ---- workspace carve-up (all 256B aligned) ----
  char* ws = (char*)d_ws;
  size_t off = 0;
  auto carve = [&](size_t bytes) -> float* {
    float* p = (float*)(ws + off);
    off = (off + bytes + 255) & ~(size_t)255;
    return p;
  };
  const size_t nodeF   = (size_t)n_nodes * sizeof(float);
  const size_t nodeH   = (size_t)n_nodes * HIDDEN * sizeof(float);
  float* dinv   = carve(nodeF);                        // degree -> dinv (in place)
  float* xw     = carve(nodeH);                        // x @ W  (per layer)
  float* agg    = carve(nodeH);                        // scattered aggregate
  float* hbuf   = carve(nodeH);                        // layer output
  float* pooled = carve((size_t)n_graphs * HIDDEN * sizeof(float));
  float* cnt    = carve((size_t)n_graphs * sizeof(float));
  (void)ws_size; (void)n_in;

  const int B = 256;
  const long long elemsH = (long long)n_nodes * HIDDEN;
  const int gridNodeH  = (int)((elemsH + B - 1) / B);
  const int gridEdge16 = (int)((n_edges * 16 + B - 1) / B);
  const int gridGemm   = n_nodes / 32;                 // 100000 % 32 == 0

  // ---- degree / normalization ----
  fill_kernel<<<(n_nodes + B - 1) / B, B, 0, stream>>>(dinv, 1.0f, n_nodes);
  deg_kernel<<<(int)((n_edges + B - 1) / B), B, 0, stream>>>(dst, dinv, (int)n_edges);
  dinv_kernel<<<(n_nodes + B - 1) / B, B, 0, stream>>>(dinv, n_nodes);

  // ---- layer 1: h = relu(Ahat @ (x W1) + b1) ----
  wmma_gemm_kernel<128><<<gridGemm, B, 0, stream>>>(x, W1, xw, n_nodes);
  fill_kernel<<<gridNodeH, B, 0, stream>>>(agg, 0.0f, elemsH);
  scatter_kernel<<<gridEdge16, B, 0, stream>>>(xw, src, dst, dinv, agg, n_edges);
  fuse_kernel<<<gridNodeH, B, 0, stream>>>(agg, xw, dinv, b1, hbuf, n_nodes, 1);

  // ---- layer 2: h = Ahat @ (h W2) + b2 ----
  wmma_gemm_kernel<64><<<gridGemm, B, 0, stream>>>(hbuf, W2, xw, n_nodes);
  fill_kernel<<<gridNodeH, B, 0, stream>>>(agg, 0.0f, elemsH);
  scatter_kernel<<<gridEdge16, B, 0, stream>>>(xw, src, dst, dinv, agg, n_edges);
  fuse_kernel<<<gridNodeH, B, 0, stream>>>(agg, xw, dinv, b2, hbuf, n_nodes, 0);

  // ---- global mean pool + linear head ----
  fill_kernel<<<(n_graphs * HIDDEN + B - 1) / B, B, 0, stream>>>(pooled, 0.0f, n_graphs * HIDDEN);
  fill_kernel<<<1, B, 0, stream>>>(cnt, 0.0f, n_graphs);
  pool_kernel<<<gridNodeH, B, 0, stream>>>(hbuf, batch, pooled, cnt, n_nodes);
  head_kernel<<<(n_graphs * N_OUT_CLS + B - 1) / B, B, 0, stream>>>(
      pooled, cnt, Wlin, blin, out, n_graphs);
}